// GraphAttentionLayer_41162966565237
// MI455X (gfx1250) — compile-verified
//
#include <hip/hip_runtime.h>
#include <hip/hip_bf16.h>

// Problem constants (match reference)
constexpr int Bb   = 4;
constexpr int Nn   = 2048;
constexpr int DIN  = 512;
constexpr int DOUT = 512;
constexpr int Hh   = 8;
constexpr int HD   = 64;   // DOUT / H

typedef float v2f __attribute__((ext_vector_type(2)));
typedef float v8f __attribute__((ext_vector_type(8)));
typedef int   v4i __attribute__((ext_vector_type(4)));

// Pointer types for the async-to-LDS builtin (param types revealed by clang:
// arg0 = v4i in AS1 (__device__/global), arg1 = v4i in AS3 (__shared__/LDS)).
typedef __attribute__((address_space(1))) v4i* as1_v4i_ptr;
typedef __attribute__((address_space(3))) v4i* as3_v4i_ptr;

#if __has_builtin(__builtin_amdgcn_global_load_async_to_lds_b128)
#define HAS_ASYNC_LDS 1
#else
#define HAS_ASYNC_LDS 0
#endif

__device__ __forceinline__ v8f wmma_f32(v2f a, v2f b, v8f c) {
    // V_WMMA_F32_16X16X4_F32 : D(16x16 f32) = A(16x4 f32) * B(4x16 f32) + C
    return __builtin_amdgcn_wmma_f32_16x16x4_f32(
        false, a, false, b, (short)0, c, false, false);
}

// ---------------------------------------------------------------------------
// Kernel 0: bit-pack the attention mask.  int32 [B,1,N,N] -> 1 bit per entry.
// 67 MB -> 2 MB, so the 8x per-head re-read stays trivially in cache.
// ---------------------------------------------------------------------------
__global__ __launch_bounds__(256)
void mask_pack_kernel(const int* __restrict__ mask, unsigned* __restrict__ bits) {
    const size_t w = (size_t)blockIdx.x * blockDim.x + threadIdx.x;  // word id
    const int* src = mask + w * 32;
    unsigned m = 0;
#pragma unroll
    for (int i = 0; i < 32; ++i)
        m |= (src[i] != 0 ? 1u : 0u) << i;
    bits[w] = m;
}

// ---------------------------------------------------------------------------
// Kernel 1: QKV projection.  y = x @ W^T + b, stored as [B, H, N, HD] f32.
// One wave per 32x64 output tile (2x4 sub-tiles of 16x16): 8 WMMAs per
// k-step fed by 6 b64 loads -> 0.75 loads/WMMA instead of 2.
// ---------------------------------------------------------------------------
__global__ __launch_bounds__(256)
void qkv_proj_kernel(const float* __restrict__ x,
                     const float* __restrict__ Wq, const float* __restrict__ bq,
                     const float* __restrict__ Wk, const float* __restrict__ bk,
                     const float* __restrict__ Wv, const float* __restrict__ bv,
                     float* __restrict__ qkv) {
    const int lane = threadIdx.x & 31;
    const int wave = threadIdx.x >> 5;
    const int tile = blockIdx.x * (blockDim.x >> 5) + wave;

    constexpr int NT = DOUT / 64;           // 8  (64-col supertiles)
    constexpr int MT = (Bb * Nn) / 32;      // 256 (32-row supertiles)
    const int mat = tile / (MT * NT);       // 0=Q, 1=K, 2=V
    const int rem = tile % (MT * NT);
    const int mt  = rem / NT;
    const int nt  = rem % NT;

    const float* W    = (mat == 0) ? Wq : (mat == 1) ? Wk : Wv;
    const float* bias = (mat == 0) ? bq : (mat == 1) ? bk : bv;

    const int half = lane >> 4;
    const int lr   = lane & 15;

    const size_t arow0 = (size_t)(mt * 32 + lr) * DIN;
    const size_t arow1 = arow0 + (size_t)16 * DIN;
    size_t brow[4];
#pragma unroll
    for (int s = 0; s < 4; ++s)
        brow[s] = (size_t)(nt * 64 + s * 16 + lr) * DIN;

    v8f acc[2][4] = {};
    for (int k = 0; k < DIN; k += 4) {
        const int ka = k + 2 * half;        // even -> 8B aligned
        v2f a0 = *(const v2f*)(x + arow0 + ka);
        v2f a1 = *(const v2f*)(x + arow1 + ka);
#pragma unroll
        for (int s = 0; s < 4; ++s) {
            v2f bb = *(const v2f*)(W + brow[s] + ka);
            acc[0][s] = wmma_f32(a0, bb, acc[0][s]);
            acc[1][s] = wmma_f32(a1, bb, acc[1][s]);
        }
    }

    float* dst = qkv + (size_t)mat * Bb * Nn * DOUT;
#pragma unroll
    for (int s = 0; s < 4; ++s) {
        const int gcol = nt * 64 + s * 16 + lr;     // 0 .. DOUT-1
        const float bn = bias[gcol];
        const int h = gcol / HD, d = gcol % HD;
#pragma unroll
        for (int mi = 0; mi < 2; ++mi) {
#pragma unroll
            for (int v = 0; v < 8; ++v) {
                const int grow = mt * 32 + mi * 16 + v + 8 * half;
                const int b = grow / Nn, n = grow % Nn;
                dst[(((size_t)b * Hh + h) * Nn + n) * HD + d] = acc[mi][s][v] + bn;
            }
        }
    }
}

// ---------------------------------------------------------------------------
// Kernel 2: Flash attention.  Block = 128 threads (4 waves) = 64 query rows.
// K/V staged per 64-key chunk into LDS (async-to-LDS when available),
// rows padded to 68 floats to kill the 16-way bank conflict on K reads.
// ---------------------------------------------------------------------------
constexpr int KC   = 64;   // keys per LDS chunk
constexpr int KLDP = 68;   // padded LDS row stride (272B: 16B aligned, bank-shifted)
constexpr int PLDP = 18;   // padded P-bounce row stride

__global__ __launch_bounds__(128)
void flash_attn_kernel(const float* __restrict__ Q,
                       const float* __restrict__ K,
                       const float* __restrict__ V,
                       const unsigned* __restrict__ mbits,
                       float* __restrict__ out) {
    __shared__ float kS[KC * KLDP];
    __shared__ float vS[KC * KLDP];
    __shared__ float pS[4][16 * PLDP];

    const int lane = threadIdx.x & 31;
    const int wave = threadIdx.x >> 5;
    const int half = lane >> 4;
    const int lr   = lane & 15;
    const int b = blockIdx.z, h = blockIdx.y;
    const int q0 = blockIdx.x * 64 + wave * 16;
    const size_t bh = ((size_t)b * Hh + h) * Nn;

    // Q tile (16 x 64) in registers, WMMA A-layout: 16 chunks of 16x4.
    v2f qa[16];
    {
        const float* Qrow = Q + (bh + q0 + lr) * HD;
#pragma unroll
        for (int kc = 0; kc < 16; ++kc)
            qa[kc] = *(const v2f*)(Qrow + kc * 4 + 2 * half);
    }

    v8f ctx[4] = {};
    float mrun[8], lrun[8];
#pragma unroll
    for (int v = 0; v < 8; ++v) { mrun[v] = -3.0e38f; lrun[v] = 0.0f; }

    const int tid = threadIdx.x;
    const int ldr = tid >> 1;               // 0..63 : K/V row staged by this thread
    const int ldc = (tid & 1) * 32;         // col base

    for (int kt = 0; kt < Nn; kt += KC) {
        __syncthreads();                    // all waves done with previous chunk
        {
            const float* Kg = K + (bh + kt + ldr) * HD + ldc;
            const float* Vg = V + (bh + kt + ldr) * HD + ldc;
            float* kDst = kS + ldr * KLDP + ldc;
            float* vDst = vS + ldr * KLDP + ldc;
#if HAS_ASYNC_LDS
#pragma unroll
            for (int c = 0; c < 8; ++c) {
                __builtin_amdgcn_global_load_async_to_lds_b128(
                    (as1_v4i_ptr)(Kg + c * 4), (as3_v4i_ptr)(kDst + c * 4), 0, 0);
                __builtin_amdgcn_global_load_async_to_lds_b128(
                    (as1_v4i_ptr)(Vg + c * 4), (as3_v4i_ptr)(vDst + c * 4), 0, 0);
            }
#if __has_builtin(__builtin_amdgcn_s_wait_asynccnt)
            __builtin_amdgcn_s_wait_asynccnt(0);
#else
            asm volatile("s_wait_asynccnt 0x0" ::: "memory");
#endif
#else
#pragma unroll
            for (int c = 0; c < 8; ++c) {
                *(float4*)(kDst + c * 4) = *(const float4*)(Kg + c * 4);
                *(float4*)(vDst + c * 4) = *(const float4*)(Vg + c * 4);
            }
#endif
            if (kt + KC < Nn) {             // hint next chunk toward L2
                __builtin_prefetch(K + (bh + kt + KC + ldr) * HD + ldc, 0, 1);
                __builtin_prefetch(V + (bh + kt + KC + ldr) * HD + ldc, 0, 1);
            }
        }
        __syncthreads();

        for (int j = 0; j < KC; j += 16) {
            // ---- S = Q @ K^T (16x16) ----
            v8f s = {};
#pragma unroll
            for (int kc = 0; kc < 16; ++kc) {
                v2f bb = *(const v2f*)(kS + (j + lr) * KLDP + kc * 4 + 2 * half);
                s = wmma_f32(qa[kc], bb, s);
            }

            // ---- scale + bit-packed mask ----
            const int kcol = kt + j + lr;
#pragma unroll
            for (int v = 0; v < 8; ++v) {
                const int qrow = q0 + v + 8 * half;
                const unsigned w =
                    mbits[((size_t)b * Nn + qrow) * (Nn / 32) + (kcol >> 5)];
                const bool keep = (w >> (kcol & 31)) & 1u;
                s[v] = keep ? s[v] * 0.125f : -1.0e9f;   // 1/sqrt(64)
            }

            // ---- online softmax ----
            float mnew[8], scale[8];
#pragma unroll
            for (int v = 0; v < 8; ++v) {
                float r = s[v];
                r = fmaxf(r, __shfl_xor(r, 1, 32));
                r = fmaxf(r, __shfl_xor(r, 2, 32));
                r = fmaxf(r, __shfl_xor(r, 4, 32));
                r = fmaxf(r, __shfl_xor(r, 8, 32));
                mnew[v]  = fmaxf(mrun[v], r);
                scale[v] = __expf(mrun[v] - mnew[v]);
                mrun[v]  = mnew[v];
            }
            v8f p;
#pragma unroll
            for (int v = 0; v < 8; ++v) {
                float pv = __expf(s[v] - mnew[v]);
                p[v] = pv;
                float rs = pv;
                rs += __shfl_xor(rs, 1, 32);
                rs += __shfl_xor(rs, 2, 32);
                rs += __shfl_xor(rs, 4, 32);
                rs += __shfl_xor(rs, 8, 32);
                lrun[v] = lrun[v] * scale[v] + rs;
            }
#pragma unroll
            for (int t = 0; t < 4; ++t)
#pragma unroll
                for (int v = 0; v < 8; ++v)
                    ctx[t][v] *= scale[v];

            // ---- P: C-layout -> A-layout via per-wave LDS bounce ----
            float* pb = pS[wave];
#pragma unroll
            for (int v = 0; v < 8; ++v)
                pb[(v + 8 * half) * PLDP + lr] = p[v];
            // same-wave DS ops are in-order; compiler inserts s_wait_dscnt
            v2f pa[4];
#pragma unroll
            for (int kc = 0; kc < 4; ++kc)
                pa[kc] = *(const v2f*)(pb + lr * PLDP + kc * 4 + 2 * half);

            // ---- ctx += P(16x16) @ V(16xHD) ----
#pragma unroll
            for (int t = 0; t < 4; ++t) {
                v8f c = ctx[t];
#pragma unroll
                for (int kc = 0; kc < 4; ++kc) {
                    const int krow = j + kc * 4 + 2 * half;
                    v2f bb;
                    bb.x = vS[krow * KLDP + t * 16 + lr];
                    bb.y = vS[(krow + 1) * KLDP + t * 16 + lr];
                    c = wmma_f32(pa[kc], bb, c);
                }
                ctx[t] = c;
            }
        }
    }

    // ---- finalize: divide by l, ELU, store [B, N, DOUT] ----
#pragma unroll
    for (int t = 0; t < 4; ++t) {
#pragma unroll
        for (int v = 0; v < 8; ++v) {
            float y = ctx[t][v] / lrun[v];
            y = (y > 0.0f) ? y : (__expf(y) - 1.0f);
            const int qrow = q0 + v + 8 * half;
            out[((size_t)b * Nn + qrow) * DOUT + h * HD + t * 16 + lr] = y;
        }
    }
}

// ---------------------------------------------------------------------------
extern "C" void kernel_launch(void* const* d_in, const int* in_sizes, int n_in,
                              void* d_out, int out_size, void* d_ws, size_t ws_size,
                              hipStream_t stream) {
    const float* x  = (const float*)d_in[0];
    const float* Wq = (const float*)d_in[1];
    const float* bq = (const float*)d_in[2];
    const float* Wk = (const float*)d_in[3];
    const float* bk = (const float*)d_in[4];
    const float* Wv = (const float*)d_in[5];
    const float* bv = (const float*)d_in[6];
    const int* mask = (const int*)d_in[7];
    float* out = (float*)d_out;

    float* qkv = (float*)d_ws;                       // [3][B,H,N,HD] = ~50.3 MB
    const size_t QSZ = (size_t)Bb * Nn * DOUT;
    unsigned* mbits = (unsigned*)(qkv + 3 * QSZ);    // +2 MB bit-packed mask

    // Kernel 0: pack mask (B*N*N/32 words)
    const int words = Bb * Nn * (Nn / 32);           // 524288
    mask_pack_kernel<<<words / 256, 256, 0, stream>>>(mask, mbits);

    // Kernel 1: 3 * 256 * 8 = 6144 wave-tiles, 8 waves/block -> 768 blocks
    const int tiles = 3 * ((Bb * Nn) / 32) * (DOUT / 64);
    qkv_proj_kernel<<<tiles / 8, 256, 0, stream>>>(x, Wq, bq, Wk, bk, Wv, bv, qkv);

    // Kernel 2: grid (N/64, H, B) x 128 threads
    dim3 g(Nn / 64, Hh, Bb);
    flash_attn_kernel<<<g, 128, 0, stream>>>(qkv, qkv + QSZ, qkv + 2 * QSZ, mbits, out);
}